// Additive_Attention_41377714929885
// MI455X (gfx1250) — compile-verified
//
#include <hip/hip_runtime.h>
#include <hip/hip_bf16.h>
#include <math.h>

#ifndef __has_builtin
#define __has_builtin(x) 0
#endif

// Problem dims (fixed by the reference)
#define B_   2
#define H_   4
#define LQ_  512
#define LKV_ 512
#define D_   64

typedef float v2f __attribute__((ext_vector_type(2)));
typedef float v8f __attribute__((ext_vector_type(8)));

__device__ __forceinline__ float fast_tanhf(float x) {
#if __has_builtin(__builtin_amdgcn_tanhf)
  return __builtin_amdgcn_tanhf(x);        // V_TANH_F32 on gfx1250
#elif __has_builtin(__builtin_amdgcn_tanh_f32)
  return __builtin_amdgcn_tanh_f32(x);
#else
  return tanhf(x);
#endif
}

// ---------------------------------------------------------------------------
// Kernel 1: qp = Q @ Wq^T (+ b_concat folded), kp = K @ Wk^T
// via V_WMMA_F32_16X16X4_F32. One wave per 16x16 output tile.
// 16 "matrices" (8 bh x {Q,K}), each 512x64x64 GEMM = 32x4 tiles -> 2048 waves.
// ---------------------------------------------------------------------------
__global__ __launch_bounds__(256) void proj_wmma_kernel(
    const float* __restrict__ Q, const float* __restrict__ K,
    const float* __restrict__ Wc,   // (64,128) row-major: cols [0,64)=Wq, [64,128)=Wk
    const float* __restrict__ bc,   // (64,)
    float* __restrict__ qp, float* __restrict__ kp)
{
  const int wid  = blockIdx.x * 8 + (threadIdx.x >> 5);  // global wave id, 0..2047
  const int mat  = wid >> 7;                             // 0..15
  const int tile = wid & 127;                            // 0..127
  const int bh   = mat >> 1;
  const int isK  = mat & 1;
  const int row0 = (tile >> 2) * 16;                     // M tile
  const int col0 = (tile & 3)  * 16;                     // N tile (e)
  const int lane = threadIdx.x & 31;
  const int l    = lane & 15;
  const int hi   = lane >> 4;                            // 0: K={0,1}, 1: K={2,3}

  const float* X     = (isK ? K : Q) + bh * (LQ_ * D_);
  const int    wofs  = isK ? D_ : 0;
  float*       Obase = (isK ? kp : qp) + bh * (LQ_ * D_);

  const float* arow = X  + (row0 + l) * D_;              // A: X rows (16x4 slab per step)
  const float* brow = Wc + (col0 + l) * (2 * D_) + wofs; // B: W^T[k,n] = W[n,k]

  v8f c = {0.f, 0.f, 0.f, 0.f, 0.f, 0.f, 0.f, 0.f};

  #pragma unroll
  for (int kk = 0; kk < D_; kk += 4) {
    const float2 av = *(const float2*)(arow + kk + 2 * hi);
    const float2 bv = *(const float2*)(brow + kk + 2 * hi);
    v2f a; a.x = av.x; a.y = av.y;
    v2f b; b.x = bv.x; b.y = bv.y;
#if __has_builtin(__builtin_amdgcn_wmma_f32_16x16x4_f32)
    // 8 args: (neg_a, A, neg_b, B, c_mod, C, reuse_a, reuse_b)
    c = __builtin_amdgcn_wmma_f32_16x16x4_f32(false, a, false, b, (short)0, c,
                                              false, false);
#else
    // Fallback so the file always compiles; histogram (wmma==0) flags this path.
    c[0] += a.x * b.x + a.y * b.y;
#endif
  }

  const int   e    = col0 + l;
  const float bias = isK ? 0.f : bc[e];   // fold b_concat into qp
  #pragma unroll
  for (int j = 0; j < 8; ++j) {
    const int q = row0 + j + hi * 8;      // C/D layout: lanes16-31 hold M=j+8
    Obase[q * D_ + e] = c[j] + bias;
  }
}

// ---------------------------------------------------------------------------
// Kernel 2: logits[q,k] = b_logit + sum_e tanh(qp'[q,e] + kp[k,e]) * w_logit[e]
// then mask (with all-false-row override) + softmax over k.
// One 128-thread block per (b,h,q) row; kp staged through LDS in 4 tiles of 128
// rows, stride padded to 65 floats (64 banks -> conflict-free).
// ---------------------------------------------------------------------------
__global__ __launch_bounds__(128) void logits_softmax_kernel(
    const float* __restrict__ qp, const float* __restrict__ kp,
    const unsigned char* __restrict__ mask,   // jnp bool, 1 byte/elem (assumed)
    const float* __restrict__ w_logit, const float* __restrict__ b_logit,
    float* __restrict__ out)
{
  __shared__ float qw[2 * D_];              // [0,64): qp row (+bias); [64,128): w_logit
  __shared__ float kt[128 * (D_ + 1)];      // 33,280 B, padded stride 65
  __shared__ float red[128];
  __shared__ int   ired[128];

  const int tid = threadIdx.x;
  const int bhq = blockIdx.x;               // 0 .. B*H*LQ-1
  const int q   = bhq & (LQ_ - 1);
  const int bh  = bhq >> 9;                 // LQ = 512
  const int b   = bh >> 2;                  // H = 4

  if (tid < D_) {
    qw[tid]      = qp[((size_t)bh * LQ_ + q) * D_ + tid];
    qw[D_ + tid] = w_logit[tid];
  }
  __syncthreads();

  const float blg = b_logit[0];
  const unsigned char* mrow = mask + ((size_t)b * LQ_ + q) * LKV_;

  float lg[4];
  unsigned char mloc[4];
  int myOr = 0;

  for (int t = 0; t < 4; ++t) {
    // Coalesced cooperative load: 128 rows x 64 floats = 2048 float4 chunks.
    const float* src = kp + ((size_t)bh * LKV_ + t * 128) * D_;
    #pragma unroll
    for (int i = 0; i < 16; ++i) {
      const int c4  = i * 128 + tid;        // consecutive lanes -> consecutive float4
      const int r   = c4 >> 4;
      const int col = (c4 & 15) * 4;
      const float4 v = *(const float4*)(src + r * D_ + col);
      float* dst = &kt[r * (D_ + 1) + col];
      dst[0] = v.x; dst[1] = v.y; dst[2] = v.z; dst[3] = v.w;
    }
    __syncthreads();

    const int k = t * 128 + tid;
    const float* krow = &kt[tid * (D_ + 1)];
    float acc = blg;
    #pragma unroll 8
    for (int e = 0; e < D_; ++e)
      acc += fast_tanhf(qw[e] + krow[e]) * qw[D_ + e];
    lg[t]   = acc;
    mloc[t] = mrow[k];
    myOr   |= mloc[t];
    __syncthreads();                        // before kt is overwritten
  }

  // Row-wide "any mask true" (reference un-masks rows whose mask is all False).
  ired[tid] = myOr;
  __syncthreads();
  for (int s = 64; s > 0; s >>= 1) {
    if (tid < s) ired[tid] |= ired[tid + s];
    __syncthreads();
  }
  const bool anyMask = (ired[0] != 0);
  __syncthreads();

  const float NEG_INF = -__builtin_inff();
  float vmax = NEG_INF;
  #pragma unroll
  for (int t = 0; t < 4; ++t) {
    if (anyMask && !mloc[t]) lg[t] = NEG_INF;
    vmax = fmaxf(vmax, lg[t]);
  }

  red[tid] = vmax;
  __syncthreads();
  for (int s = 64; s > 0; s >>= 1) {
    if (tid < s) red[tid] = fmaxf(red[tid], red[tid + s]);
    __syncthreads();
  }
  const float rmax = red[0];
  __syncthreads();

  float ex[4], vsum = 0.f;
  #pragma unroll
  for (int t = 0; t < 4; ++t) { ex[t] = __expf(lg[t] - rmax); vsum += ex[t]; }

  red[tid] = vsum;
  __syncthreads();
  for (int s = 64; s > 0; s >>= 1) {
    if (tid < s) red[tid] += red[tid + s];
    __syncthreads();
  }
  const float inv = 1.0f / red[0];

  float* orow = out + (size_t)bhq * LKV_;
  #pragma unroll
  for (int t = 0; t < 4; ++t) orow[t * 128 + tid] = ex[t] * inv;
}

// ---------------------------------------------------------------------------
extern "C" void kernel_launch(void* const* d_in, const int* in_sizes, int n_in,
                              void* d_out, int out_size, void* d_ws, size_t ws_size,
                              hipStream_t stream) {
  (void)in_sizes; (void)n_in; (void)out_size; (void)ws_size;
  const float*         Q    = (const float*)d_in[0];
  const float*         K    = (const float*)d_in[1];
  /* values d_in[2] unused by the reference output */
  const unsigned char* mask = (const unsigned char*)d_in[3];
  const float*         Wc   = (const float*)d_in[4];
  const float*         bc   = (const float*)d_in[5];
  const float*         wl   = (const float*)d_in[6];
  const float*         bl   = (const float*)d_in[7];

  float* qp = (float*)d_ws;                          // 8*512*64 floats = 1 MB
  float* kp = qp + (size_t)B_ * H_ * LQ_ * D_;       // + 1 MB (ws >= 2 MB)

  proj_wmma_kernel<<<256, 256, 0, stream>>>(Q, K, Wc, bc, qp, kp);
  logits_softmax_kernel<<<B_ * H_ * LQ_, 128, 0, stream>>>(
      qp, kp, mask, wl, bl, (float*)d_out);
}